// Encoder_87814901334660
// MI455X (gfx1250) — compile-verified
//
#include <hip/hip_runtime.h>

typedef __attribute__((ext_vector_type(2))) float v2f;
typedef __attribute__((ext_vector_type(8))) float v8f;

#define FEATS 128
#define WAVES 4
#define MT 2                                    // M-tiles (16 rows each) per wave
#define ROWS_PER_WAVE (MT * 16)                 // 32
#define ROWS_PER_BLOCK (WAVES * ROWS_PER_WAVE)  // 128
#define HPAD 132                                // row stride -> distinct LDS banks
#define LDS_BYTES (WAVES * ROWS_PER_WAVE * HPAD * 4)

__device__ inline v8f wmma4(v2f a, v2f b, v8f c) {
  return __builtin_amdgcn_wmma_f32_16x16x4_f32(
      /*neg_a=*/false, a, /*neg_b=*/false, b,
      /*c_mod=*/(short)0, c, /*reuse_a=*/false, /*reuse_b=*/false);
}

// ---------------- zero fill ----------------
__global__ __launch_bounds__(256) void zero_f32(float* __restrict__ p, long long n) {
  long long i = (long long)blockIdx.x * blockDim.x + threadIdx.x;
  long long stride = (long long)gridDim.x * blockDim.x;
  for (; i < n; i += stride) p[i] = 0.0f;
}

// ---------------- in-degree (as float, reused by both layers) ----------------
__global__ __launch_bounds__(256) void degree_kernel(const int* __restrict__ dst,
                                                     float* __restrict__ deg, int n_edges) {
  int e = blockIdx.x * blockDim.x + threadIdx.x;
  if (e < n_edges) atomicAdd(&deg[dst[e]], 1.0f);
}

// ---------------- W swizzle: Wsw[q*128+n] = (W[2q][n], W[2q+1][n]) ----------------
// Makes a B fragment (two K-consecutive values at fixed n) one contiguous b64.
__global__ __launch_bounds__(256) void wswizzle(const float* __restrict__ W,
                                                float2* __restrict__ Wsw) {
  int t = blockIdx.x * blockDim.x + threadIdx.x;  // 64*128 = 8192 threads
  int q = t >> 7, n = t & 127;
  Wsw[t] = make_float2(W[(2 * q) * FEATS + n], W[(2 * q + 1) * FEATS + n]);
}

// ---------------- edge scatter: agg[dst] += x[src] ----------------
// one wave per edge; lane l handles features 4l..4l+3 (float4 gather, 4 f32 atomics)
__global__ __launch_bounds__(256) void scatter_kernel(const float* __restrict__ x,
                                                      const int* __restrict__ src,
                                                      const int* __restrict__ dst,
                                                      float* __restrict__ agg, int n_edges) {
  long long t = (long long)blockIdx.x * blockDim.x + threadIdx.x;
  int e = (int)(t >> 5);
  if (e >= n_edges) return;
  int l = (int)(t & 31);
  const float4* xr = (const float4*)(x + (long long)src[e] * FEATS);
  float4 v = xr[l];
  float* ar = agg + (long long)dst[e] * FEATS + l * 4;
  atomicAdd(ar + 0, v.x);
  atomicAdd(ar + 1, v.y);
  atomicAdd(ar + 2, v.z);
  atomicAdd(ar + 3, v.w);
}

// ---------------- fused: h = x + agg/max(deg,1); out = act(h @ W + b) ----------------
template <bool GUARD>
__device__ inline void stage_tile(float* __restrict__ sHw, const float* __restrict__ x,
                                  const float* __restrict__ agg, const float* __restrict__ deg,
                                  int row_base, int lane, int n_nodes) {
  for (int t = lane; t < ROWS_PER_WAVE * (FEATS / 4); t += 32) {
    int r = t >> 5;
    int c = (t & 31) * 4;
    int node = row_base + r;
    float4 h = make_float4(0.f, 0.f, 0.f, 0.f);
    if (!GUARD || node < n_nodes) {
      float inv = 1.0f / fmaxf(deg[node], 1.0f);
      long long o = (long long)node * FEATS + c;
      float4 xv = *(const float4*)(x + o);
      float4 av = *(const float4*)(agg + o);
      h = make_float4(xv.x + av.x * inv, xv.y + av.y * inv,
                      xv.z + av.z * inv, xv.w + av.w * inv);
    }
    float* d = sHw + r * HPAD + c;
    d[0] = h.x; d[1] = h.y; d[2] = h.z; d[3] = h.w;
  }
}

template <bool GUARD>
__device__ inline void store_tile(const v8f (&acc)[MT][8], const float* __restrict__ bias,
                                  float* __restrict__ out, int row_base, int m16, int half,
                                  int n_nodes, int do_relu) {
#pragma unroll
  for (int mt = 0; mt < MT; ++mt) {
#pragma unroll
    for (int j = 0; j < 8; ++j) {
      int n = j * 16 + m16;
      float bv = bias[n];
#pragma unroll
      for (int r = 0; r < 8; ++r) {
        int node = row_base + mt * 16 + r + half * 8;
        if (GUARD && node >= n_nodes) continue;
        float v = acc[mt][j][r] + bv;
        if (do_relu) v = fmaxf(v, 0.0f);
        out[(long long)node * FEATS + n] = v;
      }
    }
  }
}

__global__ __launch_bounds__(WAVES * 32)
void gin_mlp_wmma(const float* __restrict__ x, const float* __restrict__ agg,
                  const float* __restrict__ deg, const float2* __restrict__ Wsw,
                  const float* __restrict__ bias, float* __restrict__ out,
                  int n_nodes, int do_relu) {
  extern __shared__ float sH[];  // [WAVES][ROWS_PER_WAVE][HPAD]

  const int wave = threadIdx.x >> 5;
  const int lane = threadIdx.x & 31;
  float* sHw = sH + wave * ROWS_PER_WAVE * HPAD;
  const int row_base = blockIdx.x * ROWS_PER_BLOCK + wave * ROWS_PER_WAVE;
  const bool full = (row_base + ROWS_PER_WAVE) <= n_nodes;  // wave-uniform

  if (full) stage_tile<false>(sHw, x, agg, deg, row_base, lane, n_nodes);
  else      stage_tile<true >(sHw, x, agg, deg, row_base, lane, n_nodes);
  __syncthreads();

  const int m16 = lane & 15;   // A row / B,C column within 16-tile
  const int half = lane >> 4;  // lanes 0-15: K-pair {0,1}; lanes 16-31: {2,3}

  v8f acc[MT][8] = {};

  for (int k4 = 0; k4 < FEATS; k4 += 4) {
    // A fragments for both M-tiles (b64 from LDS, bank-conflict-free via HPAD)
    const float* ar0 = sHw + m16 * HPAD + k4 + 2 * half;
    const float* ar1 = sHw + (16 + m16) * HPAD + k4 + 2 * half;
    v2f a0, a1;
    a0.x = ar0[0]; a0.y = ar0[1];
    a1.x = ar1[0]; a1.y = ar1[1];

    const int q = (k4 >> 1) + half;  // K-pair index into swizzled W
#pragma unroll
    for (int j = 0; j < 8; ++j) {
      int n = j * 16 + m16;
      float2 bw = Wsw[q * FEATS + n];  // single global_load_b64, coalesced
      v2f b;
      b.x = bw.x; b.y = bw.y;
      acc[0][j] = wmma4(a0, b, acc[0][j]);  // B reused across both M-tiles
      acc[1][j] = wmma4(a1, b, acc[1][j]);
    }
  }

  if (full) store_tile<false>(acc, bias, out, row_base, m16, half, n_nodes, do_relu);
  else      store_tile<true >(acc, bias, out, row_base, m16, half, n_nodes, do_relu);
}

extern "C" void kernel_launch(void* const* d_in, const int* in_sizes, int n_in,
                              void* d_out, int out_size, void* d_ws, size_t ws_size,
                              hipStream_t stream) {
  const float* x  = (const float*)d_in[0];
  const int*   src = (const int*)d_in[1];
  const int*   dst = (const int*)d_in[2];
  const float* W1 = (const float*)d_in[3];
  const float* b1 = (const float*)d_in[4];
  const float* W2 = (const float*)d_in[5];
  const float* b2 = (const float*)d_in[6];
  float* out = (float*)d_out;

  const int n_nodes = in_sizes[0] / FEATS;
  const int n_edges = in_sizes[1];
  const long long aggN = (long long)n_nodes * FEATS;

  // Workspace: agg[N*128] | h1[N*128] | deg[N] | Wsw1[64*128] | Wsw2[64*128]
  float* agg = (float*)d_ws;
  float* h1  = agg + aggN;
  float* deg = h1 + aggN;
  float2* Wsw1 = (float2*)(deg + n_nodes);
  float2* Wsw2 = Wsw1 + (FEATS / 2) * FEATS;

  dim3 blk(256);
  const long long sthreads = (long long)n_edges * 32;
  const int sblocks = (int)((sthreads + 255) / 256);
  const int gblocks = (n_nodes + ROWS_PER_BLOCK - 1) / ROWS_PER_BLOCK;

  // graph-invariant prep: degree + weight swizzles
  zero_f32<<<256, blk, 0, stream>>>(deg, n_nodes);
  degree_kernel<<<(n_edges + 255) / 256, blk, 0, stream>>>(dst, deg, n_edges);
  wswizzle<<<(FEATS / 2) * FEATS / 256, blk, 0, stream>>>(W1, Wsw1);
  wswizzle<<<(FEATS / 2) * FEATS / 256, blk, 0, stream>>>(W2, Wsw2);

  // ---- layer 1 ----
  zero_f32<<<2048, blk, 0, stream>>>(agg, aggN);
  scatter_kernel<<<sblocks, blk, 0, stream>>>(x, src, dst, agg, n_edges);
  gin_mlp_wmma<<<gblocks, dim3(WAVES * 32), LDS_BYTES, stream>>>(x, agg, deg, Wsw1, b1, h1,
                                                                 n_nodes, /*relu=*/1);

  // ---- layer 2 ----
  zero_f32<<<2048, blk, 0, stream>>>(agg, aggN);
  scatter_kernel<<<sblocks, blk, 0, stream>>>(h1, src, dst, agg, n_edges);
  gin_mlp_wmma<<<gblocks, dim3(WAVES * 32), LDS_BYTES, stream>>>(h1, agg, deg, Wsw2, b2, out,
                                                                 n_nodes, /*relu=*/0);
}